// LengthRegulator_13056700580540
// MI455X (gfx1250) — compile-verified
//
#include <hip/hip_runtime.h>

// LengthRegulator for MI455X (gfx1250, wave32).
// B=32, T=1024, D=384, MAX_LEN=4096. Pure data movement:
//   ~192 MiB NT output stores + ~48 MiB x reads (L2-resident reuse)
//   -> HBM-bandwidth bound, ~10.5 us floor at 23.3 TB/s. No FLOPs -> no WMMA.

#define BATCH    32
#define TLEN     1024
#define DDIM     384
#define MAXLEN   4096
#define ROWS_PER_BLOCK 4   // 4 output frames per block
#define LANES_PER_ROW  96  // 96 lanes * float4 = 384 floats = one row

typedef float v4f __attribute__((ext_vector_type(4)));

// -------- Kernel 1: per-batch inclusive scan of durations (LDS Hillis-Steele)
__global__ __launch_bounds__(TLEN)
void lr_scan_kernel(const int* __restrict__ dur, int* __restrict__ cum) {
    __shared__ int s[TLEN];
    const int b = blockIdx.x;
    const int t = threadIdx.x;
    s[t] = dur[b * TLEN + t];
    __syncthreads();
#pragma unroll
    for (int off = 1; off < TLEN; off <<= 1) {
        int v = (t >= off) ? s[t - off] : 0;
        __syncthreads();
        s[t] += v;
        __syncthreads();
    }
    cum[b * TLEN + t] = s[t];
}

// -------- Kernel 2: per-output-frame searchsorted + 1536B row copy
// block = (96, 4): each y-slice handles one output frame; lane x==0 of the
// slice does the binary search, result broadcast through LDS.
__global__ __launch_bounds__(LANES_PER_ROW * ROWS_PER_BLOCK)
void lr_gather_kernel(const float* __restrict__ x,
                      const int* __restrict__ cum,
                      float* __restrict__ out) {
    __shared__ int s_idx[ROWS_PER_BLOCK];

    const int r = blockIdx.x * ROWS_PER_BLOCK + threadIdx.y; // global frame id
    const int b = r >> 12;            // r / MAXLEN   (MAXLEN == 4096)
    const int j = r & (MAXLEN - 1);   // r % MAXLEN
    const int* __restrict__ c = cum + b * TLEN;

    if (threadIdx.x == 0) {
        __builtin_prefetch(c, 0, 1);      // global_prefetch_b8 on gfx1250
        const int total = c[TLEN - 1];
        int idx = -1;
        if (j < total) {
            // searchsorted(c, j, side='right'): first i with c[i] > j.
            // Answer space is [0, TLEN] (1025 values) -> up to 11 guarded
            // iterations; guard also prevents reading c[TLEN].
            int lo = 0, hi = TLEN;
            while (lo < hi) {
                const int mid = (lo + hi) >> 1;
                if (c[mid] <= j) lo = mid + 1; else hi = mid;
            }
            idx = lo;  // < TLEN guaranteed since c[TLEN-1] = total > j
        }
        s_idx[threadIdx.y] = idx;
    }
    __syncthreads();

    const int idx = s_idx[threadIdx.y];
    v4f v = (v4f)(0.0f, 0.0f, 0.0f, 0.0f);
    if (idx >= 0) {
        const v4f* __restrict__ src =
            (const v4f*)(x + ((size_t)b * TLEN + (size_t)idx) * DDIM);
        v = src[threadIdx.x];                        // global_load_b128
    }
    v4f* __restrict__ dst =
        (v4f*)(out + ((size_t)b * MAXLEN + (size_t)j) * DDIM);
    // Output is write-once / never re-read: non-temporal store keeps the
    // 192 MiB write stream from evicting x (48 MiB, reused ~3x) out of L2.
    __builtin_nontemporal_store(v, &dst[threadIdx.x]); // global_store_b128 nt
}

extern "C" void kernel_launch(void* const* d_in, const int* in_sizes, int n_in,
                              void* d_out, int out_size, void* d_ws, size_t ws_size,
                              hipStream_t stream) {
    const float* x   = (const float*)d_in[0];   // [B, T, D] float32
    const int*   dur = (const int*)d_in[1];     // [B, T] int32
    // d_in[2] is max_len scalar == 4096 (compile-time constant here)
    float* out = (float*)d_out;                 // [B, MAXLEN, D] float32
    int*   cum = (int*)d_ws;                    // [B, T] int32 scratch (128 KB)

    lr_scan_kernel<<<BATCH, TLEN, 0, stream>>>(dur, cum);

    dim3 block(LANES_PER_ROW, ROWS_PER_BLOCK);
    dim3 grid((BATCH * MAXLEN) / ROWS_PER_BLOCK);
    lr_gather_kernel<<<grid, block, 0, stream>>>(x, cum, out);
}